// StyleLoss_26190710571731
// MI455X (gfx1250) — compile-verified
//
#include <hip/hip_runtime.h>
#include <hip/hip_bf16.h>

typedef __attribute__((ext_vector_type(2))) float v2f;
typedef __attribute__((ext_vector_type(4))) float v4f;
typedef __attribute__((ext_vector_type(8))) float v8f;

// ---------------------------------------------------------------------------
// Kernel 1: G = (feats @ feats^T) / 2^20   via V_WMMA_F32_16X16X4_F32
// feats = x viewed as 512 x 4096 fp32.  G is 512 x 512 fp32 in workspace.
// One wave per 16x16 output tile; 8 waves per block; 128 blocks = 1024 tiles.
// ---------------------------------------------------------------------------
__global__ void syrk_wmma_f32(const float* __restrict__ feats,
                              float* __restrict__ G) {
    const int wave = threadIdx.x >> 5;
    const int lane = threadIdx.x & 31;
    const int tile = blockIdx.x * 8 + wave;     // 0..1023
    const int ti   = tile >> 5;                 // row tile 0..31
    const int tj   = tile & 31;                 // col tile 0..31

    // A fragment (16x4 fp32, 2 VGPRs): lane&15 = M, lane>>4 selects K pair.
    // B fragment (4x16 fp32) mirrors A with N in place of M -> same loader,
    // base row = column-tile row of feats (B = feats^T tile).
    const int m      = lane & 15;
    const int kpair  = (lane >> 4) << 1;        // 0 or 2
    const float* arow = feats + (size_t)(ti * 16 + m) * 4096 + kpair;
    const float* brow = feats + (size_t)(tj * 16 + m) * 4096 + kpair;

    v8f acc = {};
    #pragma unroll 8
    for (int k = 0; k < 4096; k += 4) {
        v2f a = *reinterpret_cast<const v2f*>(arow + k);
        v2f b = *reinterpret_cast<const v2f*>(brow + k);
        // 8 args: (neg_a, A, neg_b, B, c_mod, C, reuse_a, reuse_b)
        acc = __builtin_amdgcn_wmma_f32_16x16x4_f32(
            false, a, false, b, (short)0, acc, false, false);
    }

    // D layout: lane&15 = N; VGPR v holds M = v + 8*(lane>>4).
    const int n  = lane & 15;
    const int mb = (lane >> 4) * 8;
    float* gout = G + (size_t)(ti * 16 + mb) * 512 + tj * 16 + n;
    const float scale = 1.0f / 1048576.0f;      // b*local_ch*h*w = 256*4096
    #pragma unroll
    for (int v = 0; v < 8; ++v)
        gout[(size_t)v * 512] = acc[v] * scale;
}

// ---------------------------------------------------------------------------
// Kernel 2: fused gather + squared error partial reduction.
// gram[c,a,b] = G[(c+a)&511][(c+b)&511]; sum (gram - target)^2 per block.
// target is streamed once -> non-temporal float4 loads (global_load_b128 NT).
// G (1 MB) is gathered 33.5M times -> regular loads, stays hot in L2.
// ---------------------------------------------------------------------------
__global__ void loss_partial(const float* __restrict__ target,
                             const float* __restrict__ G,
                             float* __restrict__ partials) {
    const unsigned N4 = 512u * 256u * 64u;      // 8388608 float4s
    const v4f* __restrict__ t4 = reinterpret_cast<const v4f*>(target);

    float sum = 0.0f;
    const unsigned stride = gridDim.x * blockDim.x;
    for (unsigned i4 = blockIdx.x * blockDim.x + threadIdx.x; i4 < N4; i4 += stride) {
        const unsigned i = i4 << 2;             // flat element index (b % 4 == 0)
        const int b  = i & 255;
        const int a  = (i >> 8) & 255;
        const int c  = i >> 16;

        const v4f t = __builtin_nontemporal_load(t4 + i4);

        const float* __restrict__ grow = G + (((c + a) & 511) << 9);
        const int cb = c + b;
        const float g0 = grow[(cb    ) & 511];
        const float g1 = grow[(cb + 1) & 511];
        const float g2 = grow[(cb + 2) & 511];
        const float g3 = grow[(cb + 3) & 511];

        const float d0 = g0 - t.x;
        const float d1 = g1 - t.y;
        const float d2 = g2 - t.z;
        const float d3 = g3 - t.w;
        sum = fmaf(d0, d0, sum);
        sum = fmaf(d1, d1, sum);
        sum = fmaf(d2, d2, sum);
        sum = fmaf(d3, d3, sum);
    }

    // wave32 shuffle reduction, then one LDS slot per wave
    #pragma unroll
    for (int off = 16; off > 0; off >>= 1)
        sum += __shfl_xor(sum, off, 32);

    __shared__ float red[8];                    // 256 threads = 8 waves
    const int wave = threadIdx.x >> 5;
    const int lane = threadIdx.x & 31;
    if (lane == 0) red[wave] = sum;
    __syncthreads();
    if (threadIdx.x == 0) {
        float s = red[0];
        #pragma unroll
        for (int w = 1; w < 8; ++w) s += red[w];
        partials[blockIdx.x] = s;
    }
}

// ---------------------------------------------------------------------------
// Kernel 3: final reduction -> loss = sum * (1e6 / 33554432)
// ---------------------------------------------------------------------------
__global__ void loss_final(const float* __restrict__ partials, int n,
                           float* __restrict__ out) {
    float s = 0.0f;
    for (int i = threadIdx.x; i < n; i += 256) s += partials[i];

    #pragma unroll
    for (int off = 16; off > 0; off >>= 1)
        s += __shfl_xor(s, off, 32);

    __shared__ float red[8];
    const int wave = threadIdx.x >> 5;
    const int lane = threadIdx.x & 31;
    if (lane == 0) red[wave] = s;
    __syncthreads();
    if (threadIdx.x == 0) {
        float tot = red[0];
        #pragma unroll
        for (int w = 1; w < 8; ++w) tot += red[w];
        out[0] = tot * (1000000.0f / 33554432.0f);
    }
}

extern "C" void kernel_launch(void* const* d_in, const int* in_sizes, int n_in,
                              void* d_out, int out_size, void* d_ws, size_t ws_size,
                              hipStream_t stream) {
    const float* x      = (const float*)d_in[0];   // (1,512,64,64) fp32
    const float* target = (const float*)d_in[1];   // (512,256,256) fp32
    float* out = (float*)d_out;                    // scalar loss

    float* G        = (float*)d_ws;                                   // 1 MB
    float* partials = (float*)((char*)d_ws + 512 * 512 * sizeof(float));

    // 1024 tiles of 16x16, 8 waves (one tile each) per 256-thread block.
    syrk_wmma_f32<<<128, 256, 0, stream>>>(x, G);

    const int nblk = 4096;
    loss_partial<<<nblk, 256, 0, stream>>>(target, G, partials);
    loss_final<<<1, 256, 0, stream>>>(partials, nblk, out);
}